// KMeans_10033043604045
// MI455X (gfx1250) — compile-verified
//
#include <hip/hip_runtime.h>

typedef float v2f __attribute__((ext_vector_type(2)));
typedef float v8f __attribute__((ext_vector_type(8)));

typedef int v4i_ __attribute__((vector_size(16)));               // 16-byte int vector
typedef __attribute__((address_space(1))) v4i_ gv4i_t;           // global b128
typedef __attribute__((address_space(3))) v4i_ lv4i_t;           // LDS b128

#define N_PTS  100000
#define DIM    256
#define KC     1024
#define MAX_IT 8
#define TOLF   1e-4f
#define TPAD   260          // padded LDS row stride (dwords): 260 % 64 = 4 -> conflict-free
#define NTILE  (KC / 16)    // 64 column tiles of 16 centers

// ---- CDNA5 async global->LDS copy (no VGPR staging, tracked by ASYNCcnt) ----
__device__ __forceinline__ void async_copy_b128(const float* gsrc, float* ldst) {
#if __has_builtin(__builtin_amdgcn_global_load_async_to_lds_b128)
    __builtin_amdgcn_global_load_async_to_lds_b128(
        (gv4i_t*)gsrc, (lv4i_t*)ldst, 0, 0);
#else
    unsigned lds_off = (unsigned)(size_t)(lv4i_t*)ldst;
    asm volatile("global_load_async_to_lds_b128 %0, %1, off"
                 :: "v"(lds_off), "v"((unsigned long long)(size_t)gsrc)
                 : "memory");
#endif
}

__device__ __forceinline__ void wait_asynccnt0() {
#if __has_builtin(__builtin_amdgcn_s_wait_asynccnt)
    __builtin_amdgcn_s_wait_asynccnt(0);
#else
    asm volatile("s_wait_asynccnt 0x0" ::: "memory");
#endif
}

// ---------------------------------------------------------------------------
// init: copy init_centers -> working centers, reset done/shift2
// ---------------------------------------------------------------------------
__global__ void km_init_kernel(const float* __restrict__ initc,
                               float* __restrict__ centers,
                               float* __restrict__ shift2,
                               int* __restrict__ done) {
    int idx = blockIdx.x * blockDim.x + threadIdx.x;
    if (idx < KC * DIM) centers[idx] = initc[idx];
    if (idx == 0) { *shift2 = 0.0f; *done = 0; }
}

// ---------------------------------------------------------------------------
// half_csq[k] = 0.5 * sum_d centers[k][d]^2
// ---------------------------------------------------------------------------
__global__ void km_csq_kernel(const float* __restrict__ centers,
                              float* __restrict__ half_csq) {
    __shared__ float s[256];
    int tid = threadIdx.x;
    float v = centers[blockIdx.x * DIM + tid];
    s[tid] = v * v;
    __syncthreads();
    for (int off = 128; off > 0; off >>= 1) {
        if (tid < off) s[tid] += s[tid + off];
        __syncthreads();
    }
    if (tid == 0) half_csq[blockIdx.x] = 0.5f * s[0];
}

// ---------------------------------------------------------------------------
// Assignment: argmin_k ( 0.5*||c_k||^2 - x . c_k )  via V_WMMA_F32_16X16X4_F32
// Block = 8 waves = 128 rows of X. Centers tiles (16 x 256 f32 = 16KB) are
// DMA'd straight into double-buffered LDS with GLOBAL_LOAD_ASYNC_TO_LDS_B128;
// tile t+1's DMA runs underneath tile t's 64-WMMA compute.
//   A (16x4 f32): lane = {row = lane&15, K = 2*(lane>>4) + [0,1]}  -> float2
//   B (4x16 f32): lane = {col = lane&15, K = 2*(lane>>4) + [0,1]}  -> float2
//   C/D (16x16 f32): elem v -> row = v + 8*(lane>>4), col = lane&15
// ---------------------------------------------------------------------------
__global__ __launch_bounds__(256, 1)
void km_assign_kernel(const float* __restrict__ X,
                      const float* __restrict__ centers,
                      const float* __restrict__ half_csq,
                      int* __restrict__ assign) {
    __shared__ float lds[2][16 * TPAD];          // 2 x 16.25 KB tile buffers

    const int tid  = threadIdx.x;
    const int lane = tid & 31;
    const int wave = tid >> 5;
    const int r0   = (blockIdx.x * 8 + wave) * 16;
    const bool valid = (r0 < N_PTS);             // whole wave uniform (N % 16 == 0)
    const int rbase  = valid ? r0 : 0;           // clamp: keep EXEC all-1s + barriers

    const int m    = lane & 15;                  // row (A) / col (B) within tile
    const int koff = (lane >> 4) * 2;            // K sub-offset per half-wave

    // ---- load this wave's 16x256 X panel into registers (64 float2) ----
    const float* xrow = X + (size_t)(rbase + m) * DIM + koff;
    v2f a[64];
#pragma unroll
    for (int t = 0; t < 64; ++t) a[t] = *(const v2f*)(xrow + 4 * t);

    float bestd[8];
    int   besti[8];
#pragma unroll
    for (int v = 0; v < 8; ++v) { bestd[v] = 3.4e38f; besti[v] = 0; }

    // thread tid owns 4 b128 slices of each tile: slice f = tid + j*256,
    // f in [0,1024): row = f>>6, c4 = f&63
    const int srow[4] = { (tid + 0)   >> 6, (tid + 256) >> 6,
                          (tid + 512) >> 6, (tid + 768) >> 6 };
    const int sc4 = tid & 63;

    // prologue: DMA tile 0 into buffer 0
#pragma unroll
    for (int j = 0; j < 4; ++j)
        async_copy_b128(centers + (size_t)srow[j] * DIM + sc4 * 4,
                        &lds[0][srow[j] * TPAD + sc4 * 4]);

    for (int tile = 0; tile < NTILE; ++tile) {
        const float* buf = lds[tile & 1];

        wait_asynccnt0();                        // my slices of tile t landed
        __syncthreads();                         // everyone's landed; buf[(t+1)&1]
                                                 // no longer read by any wave
        if (tile < NTILE - 1) {                  // DMA tile t+1 under compute
            const float* nsrc = centers + (size_t)(tile + 1) * 16 * DIM;
            float* nbuf = lds[(tile + 1) & 1];
#pragma unroll
            for (int j = 0; j < 4; ++j)
                async_copy_b128(nsrc + (size_t)srow[j] * DIM + sc4 * 4,
                                &nbuf[srow[j] * TPAD + sc4 * 4]);
        }

        const int n0 = tile * 16;
        const float* bbase = buf + m * TPAD + koff;
        v8f acc0 = {};
        v8f acc1 = {};
#pragma unroll
        for (int t = 0; t < 64; t += 2) {
            v2f b0 = *(const v2f*)(bbase + 4 * t);
            v2f b1 = *(const v2f*)(bbase + 4 * (t + 1));
            acc0 = __builtin_amdgcn_wmma_f32_16x16x4_f32(
                false, a[t],     false, b0, (short)0, acc0, false, false);
            acc1 = __builtin_amdgcn_wmma_f32_16x16x4_f32(
                false, a[t + 1], false, b1, (short)0, acc1, false, false);
        }

        const float hc   = half_csq[n0 + m];
        const int   nidx = n0 + m;
#pragma unroll
        for (int v = 0; v < 8; ++v) {
            float d = hc - (acc0[v] + acc1[v]);
            if (d < bestd[v]) { bestd[v] = d; besti[v] = nidx; }  // first-min ties
        }
    }

    // reduce min/argmin across each 16-lane group (xor masks < 16 stay in group)
#pragma unroll
    for (int v = 0; v < 8; ++v) {
        float d = bestd[v];
        int   i = besti[v];
#pragma unroll
        for (int mask = 1; mask < 16; mask <<= 1) {
            float od = __shfl_xor(d, mask, 32);
            int   oi = __shfl_xor(i, mask, 32);
            if (od < d || (od == d && oi < i)) { d = od; i = oi; }
        }
        if (valid && m == 0) assign[r0 + v + (lane >> 4) * 8] = i;
    }
}

// ---------------------------------------------------------------------------
__global__ void km_zero_kernel(float* __restrict__ p, int n) {
    int idx = blockIdx.x * blockDim.x + threadIdx.x;
    if (idx < n) p[idx] = 0.0f;
}

// ---------------------------------------------------------------------------
// Scatter-add: sums[a] += x, counts[a] += 1.  Thread = (point, 4-dim chunk).
// ---------------------------------------------------------------------------
__global__ void km_accum_kernel(const float* __restrict__ X,
                                const int* __restrict__ assign,
                                float* __restrict__ sums,
                                float* __restrict__ counts,
                                const int* __restrict__ done) {
    if (*done) return;
    int idx = blockIdx.x * blockDim.x + threadIdx.x;
    if (idx >= N_PTS * (DIM / 4)) return;
    int i = idx >> 6;          // point index
    int c = idx & 63;          // float4 chunk
    int a = assign[i];
    float4 xv = *(const float4*)(X + (size_t)i * DIM + c * 4);
    float* dst = sums + (size_t)a * DIM + c * 4;
    atomicAdd(dst + 0, xv.x);
    atomicAdd(dst + 1, xv.y);
    atomicAdd(dst + 2, xv.z);
    atomicAdd(dst + 3, xv.w);
    if (c == 0) atomicAdd(counts + a, 1.0f);
}

// ---------------------------------------------------------------------------
// centers = counts>0 ? sums/counts : centers ; shift2 += (new-old)^2
// ---------------------------------------------------------------------------
__global__ void km_update_kernel(float* __restrict__ centers,
                                 const float* __restrict__ sums,
                                 const float* __restrict__ counts,
                                 float* __restrict__ shift2,
                                 const int* __restrict__ done) {
    if (*done) return;
    __shared__ float s[256];
    int tid = threadIdx.x;
    int idx = blockIdx.x * blockDim.x + tid;      // K*D total
    int k = idx >> 8;                             // DIM == 256
    float cnt = counts[k];
    float old = centers[idx];
    float nw  = (cnt > 0.0f) ? (sums[idx] / cnt) : old;
    centers[idx] = nw;
    float df = nw - old;
    s[tid] = df * df;
    __syncthreads();
    for (int off = 128; off > 0; off >>= 1) {
        if (tid < off) s[tid] += s[tid + off];
        __syncthreads();
    }
    if (tid == 0) atomicAdd(shift2, s[0]);
}

// ---------------------------------------------------------------------------
__global__ void km_done_kernel(float* __restrict__ shift2, int* __restrict__ done) {
    if (*shift2 < TOLF) *done = 1;
    *shift2 = 0.0f;
}

// ---------------------------------------------------------------------------
// Copy centers to output; zero the inertia slot.
// ---------------------------------------------------------------------------
__global__ void km_copyout_kernel(const float* __restrict__ centers,
                                  float* __restrict__ out) {
    int idx = blockIdx.x * blockDim.x + threadIdx.x;
    if (idx < KC * DIM) out[idx] = centers[idx];
    if (idx == 0) out[KC * DIM + N_PTS] = 0.0f;
}

// ---------------------------------------------------------------------------
// inertia = sum_i || x_i - centers[assign_i] ||^2
// ---------------------------------------------------------------------------
__global__ void km_inertia_kernel(const float* __restrict__ X,
                                  const float* __restrict__ centers,
                                  const int* __restrict__ assign,
                                  float* __restrict__ out) {
    __shared__ float s[256];
    int tid = threadIdx.x;
    int i = blockIdx.x * blockDim.x + tid;
    float part = 0.0f;
    if (i < N_PTS) {
        int a = assign[i];
        const float4* xr = (const float4*)(X + (size_t)i * DIM);
        const float4* cr = (const float4*)(centers + (size_t)a * DIM);
        for (int d = 0; d < DIM / 4; ++d) {
            float4 x = xr[d], c = cr[d];
            float dx = x.x - c.x, dy = x.y - c.y, dz = x.z - c.z, dw = x.w - c.w;
            part += dx * dx + dy * dy + dz * dz + dw * dw;
        }
    }
    s[tid] = part;
    __syncthreads();
    for (int off = 128; off > 0; off >>= 1) {
        if (tid < off) s[tid] += s[tid + off];
        __syncthreads();
    }
    if (tid == 0) atomicAdd(&out[KC * DIM + N_PTS], s[0]);
}

// ---------------------------------------------------------------------------
extern "C" void kernel_launch(void* const* d_in, const int* in_sizes, int n_in,
                              void* d_out, int out_size, void* d_ws, size_t ws_size,
                              hipStream_t stream) {
    const float* X     = (const float*)d_in[0];
    const float* initc = (const float*)d_in[1];
    float* out = (float*)d_out;

    // workspace layout (floats)
    float* centers  = (float*)d_ws;              // K*D
    float* half_csq = centers + KC * DIM;        // K
    float* sums     = half_csq + KC;             // K*D
    float* counts   = sums + KC * DIM;           // K
    float* shift2   = counts + KC;               // 1
    int*   done     = (int*)(shift2 + 1);        // 1

    int* assign = (int*)(out + KC * DIM);        // assignments live in d_out

    const int waves  = (N_PTS + 15) / 16;                 // 6250
    const int ablk   = (waves + 7) / 8;                   // 782 blocks of 8 waves
    const int kdblk  = (KC * DIM) / 256;                  // 1024
    const int acblk  = (N_PTS * (DIM / 4) + 255) / 256;   // 25000
    const int inblk  = (N_PTS + 255) / 256;               // 391

    km_init_kernel<<<kdblk, 256, 0, stream>>>(initc, centers, shift2, done);

    for (int it = 0; it < MAX_IT; ++it) {
        km_csq_kernel<<<KC, 256, 0, stream>>>(centers, half_csq);
        km_assign_kernel<<<ablk, 256, 0, stream>>>(X, centers, half_csq, assign);
        km_zero_kernel<<<kdblk, 256, 0, stream>>>(sums, KC * DIM);
        km_zero_kernel<<<(KC + 255) / 256, 256, 0, stream>>>(counts, KC);
        km_accum_kernel<<<acblk, 256, 0, stream>>>(X, assign, sums, counts, done);
        km_update_kernel<<<kdblk, 256, 0, stream>>>(centers, sums, counts, shift2, done);
        km_done_kernel<<<1, 1, 0, stream>>>(shift2, done);
    }

    km_copyout_kernel<<<kdblk, 256, 0, stream>>>(centers, out);
    km_inertia_kernel<<<inblk, 256, 0, stream>>>(X, centers, assign, out);
}